// BatchCenterLoss_72619307041174
// MI455X (gfx1250) — compile-verified
//
#include <hip/hip_runtime.h>

typedef __attribute__((ext_vector_type(2))) float v2f;
typedef __attribute__((ext_vector_type(8))) float v8f;

#define B_SZ   16384
#define D_DIM  128
#define NCLS   100
#define TMAX   24              // 16-row tiles per class dim -> class-size bound 384 (mean 164)
#define TPC    (TMAX * TMAX)

// ---------------- workspace layout (bytes) ----------------
// counts: int[128]   @ 0
// cursor: int[128]   @ 512
// offs:   int[136]   @ 1024   (NCLS+1 used)
// perm:   int[B_SZ]  @ 4096
// sqn:    f32[B_SZ]  @ 4096 + 65536
// total ~132 KB

__global__ __launch_bounds__(128) void k_init(int* counts, float* out) {
    int t = threadIdx.x;
    if (t < 128) counts[t] = 0;
    if (t == 0) out[0] = 0.0f;
}

// one wave (32 lanes) per row: sqn[row] = sum_d x[row][d]^2
__global__ __launch_bounds__(256) void k_sqnorm(const float* __restrict__ x,
                                                float* __restrict__ sqn) {
    int wave = threadIdx.x >> 5;
    int lane = threadIdx.x & 31;
    int row  = blockIdx.x * 8 + wave;
    const float* p = x + (size_t)row * D_DIM;
    float s = 0.0f;
#pragma unroll
    for (int q = 0; q < 4; ++q) {
        float v = p[lane + 32 * q];
        s += v * v;
    }
#pragma unroll
    for (int o = 16; o > 0; o >>= 1) s += __shfl_down(s, o, 32);
    if (lane == 0) sqn[row] = s;
}

__global__ __launch_bounds__(256) void k_hist(const int* __restrict__ tg,
                                              int* __restrict__ counts) {
    int i = blockIdx.x * blockDim.x + threadIdx.x;
    if (i < B_SZ) atomicAdd(&counts[tg[i]], 1);
}

__global__ void k_scan(const int* __restrict__ counts, int* __restrict__ offs,
                       int* __restrict__ cursor) {
    // single thread: 100-entry exclusive scan
    int acc = 0;
    for (int c = 0; c < NCLS; ++c) {
        offs[c] = acc;
        cursor[c] = acc;
        acc += counts[c];
    }
    offs[NCLS] = acc;
}

__global__ __launch_bounds__(256) void k_scatter(const int* __restrict__ tg,
                                                 int* __restrict__ cursor,
                                                 int* __restrict__ perm) {
    int i = blockIdx.x * blockDim.x + threadIdx.x;
    if (i < B_SZ) {
        int pos = atomicAdd(&cursor[tg[i]], 1);
        perm[pos] = i;
    }
}

// One wave per 16x16 tile of the upper triangle (ti <= tj) of one class's
// pairwise-distance matrix. Gram tile via chained V_WMMA_F32_16X16X4_F32
// over K=128 (32 steps). Off-diagonal tiles weighted 2x via the final scale.
__global__ __launch_bounds__(32) void k_tiles(const float* __restrict__ x,
                                              const int* __restrict__ offs,
                                              const int* __restrict__ perm,
                                              const float* __restrict__ sqn,
                                              float* __restrict__ out) {
    int c  = blockIdx.x / TPC;
    int t  = blockIdx.x % TPC;
    int ti = t / TMAX;
    int tj = t % TMAX;
    if (tj < ti) return;                        // symmetry: upper triangle only

    int s0 = offs[c];
    int n  = offs[c + 1] - s0;                  // class population
    if (tj * 16 >= n) return;                   // wave-uniform exit (ti <= tj)

    int lane = threadIdx.x;
    int half = lane >> 4;                       // A/B layout: K-pair {0,1} vs {2,3}
    int mrow = lane & 15;

    int ia = ti * 16 + mrow; if (ia >= n) ia = n - 1;   // clamp, mask later
    int ja = tj * 16 + mrow; if (ja >= n) ja = n - 1;
    int ga = perm[s0 + ia];
    int gb = perm[s0 + ja];

    const float* pa = x + (size_t)ga * D_DIM + half * 2;
    const float* pb = x + (size_t)gb * D_DIM + half * 2;

    float sa = sqn[ga];                         // this lane's i-row norm (row = mrow)
    float sb = sqn[gb];                         // this lane's j-row norm (row = mrow)

    v8f acc = {};
#pragma unroll
    for (int k = 0; k < D_DIM; k += 4) {
        v2f a = *(const v2f*)(pa + k);          // 8B aligned: row base 512B + half*8B
        v2f b = *(const v2f*)(pb + k);
        acc = __builtin_amdgcn_wmma_f32_16x16x4_f32(
            /*neg_a=*/false, a, /*neg_b=*/false, b,
            /*c_mod=*/(short)0, acc, /*reuse_a=*/false, /*reuse_b=*/false);
    }

    // D layout: lane N = lane&15; VGPR r holds M = r (lanes 0-15) / r+8 (lanes 16-31).
    // Row-norm for M = r + 8*half lives in lane (r + 8*half) (lanes m, m+16 share row m).
    int nj = tj * 16 + mrow;
    float s = 0.0f;
#pragma unroll
    for (int r = 0; r < 8; ++r) {
        int m  = r + 8 * half;
        int mi = ti * 16 + m;
        float snA = __shfl(sa, m, 32);
        if (mi < n && nj < n && mi != nj) {
            float g  = acc[r];
            float sq = snA + sb - 2.0f * g;
            sq = sq > 0.0f ? sq : 0.0f;
            s += sqrtf(sq);
        }
    }
#pragma unroll
    for (int o = 16; o > 0; o >>= 1) s += __shfl_down(s, o, 32);

    // full-matrix sum / 2 / B  ==  sum_{ti<tj} tile / B  +  sum_{ti==tj} tile / (2B)
    float scale = (ti == tj) ? (0.5f / (float)B_SZ) : (1.0f / (float)B_SZ);
    if (lane == 0) atomicAdd(out, s * scale);
}

extern "C" void kernel_launch(void* const* d_in, const int* in_sizes, int n_in,
                              void* d_out, int out_size, void* d_ws, size_t ws_size,
                              hipStream_t stream) {
    const float* x  = (const float*)d_in[0];
    const int*   tg = (const int*)d_in[1];
    float*       out = (float*)d_out;

    char* ws = (char*)d_ws;
    int*   counts = (int*)(ws + 0);
    int*   cursor = (int*)(ws + 512);
    int*   offs   = (int*)(ws + 1024);
    int*   perm   = (int*)(ws + 4096);
    float* sqn    = (float*)(ws + 4096 + 65536);

    k_init   <<<1, 128, 0, stream>>>(counts, out);
    k_sqnorm <<<B_SZ / 8, 256, 0, stream>>>(x, sqn);
    k_hist   <<<B_SZ / 256, 256, 0, stream>>>(tg, counts);
    k_scan   <<<1, 1, 0, stream>>>(counts, offs, cursor);
    k_scatter<<<B_SZ / 256, 256, 0, stream>>>(tg, cursor, perm);
    k_tiles  <<<NCLS * TPC, 32, 0, stream>>>(x, offs, perm, sqn, out);
}